// DsaScatterGraphSafe_35811437314274
// MI455X (gfx1250) — compile-verified
//
#include <hip/hip_runtime.h>
#include <hip/hip_bf16.h>

#define AS1 __attribute__((address_space(1)))
#define AS3 __attribute__((address_space(3)))

typedef int v4i __attribute__((ext_vector_type(4)));

#if __has_builtin(__builtin_amdgcn_global_load_async_to_lds_b128)
#define HAVE_ASYNC_LDS 1
#else
#define HAVE_ASYNC_LDS 0
#endif

namespace {
constexpr int kB       = 2;
constexpr int kSQ      = 4096;
constexpr int kSKV     = 8192;
constexpr int kK       = 2048;
constexpr int kThreads = 256;
constexpr int kRowVec  = kSKV / 4;            // 2048 float4 chunks per row
constexpr int kRowPerT = kRowVec / kThreads;  // 8 float4 per thread
constexpr int kIdxVec  = kK / 4;              // 512 int4 chunks per idx row
constexpr int kIdxPerT = kIdxVec / kThreads;  // 2 int4 per thread
constexpr int kMaskWords = kSKV / 32;         // 256 bitmask words (== kThreads)
} // namespace

__global__ __launch_bounds__(kThreads)
void dsa_scatter_row_kernel(const float* __restrict__ mask,
                            const int*   __restrict__ idx,
                            const int*   __restrict__ s0p,
                            const int*   __restrict__ s1p,
                            float*       __restrict__ out)
{
    __shared__ int      lds_idx[kK];          // 8 KB: async-staged index row
    __shared__ unsigned bits[kMaskWords];     // 1 KB: one bit per column
    __shared__ unsigned flags;                // bit0 = has_sentinel, bit1 = has_real_zero

    const int row = blockIdx.x;               // 0 .. B*SQ-1
    const int b   = row / kSQ;
    const int l   = row - b * kSQ;
    const int t   = threadIdx.x;
    const int s0  = s0p[0];
    const int s1  = s1p[0];

    const float* __restrict__ src = mask + (size_t)row * kSKV;
    float*       __restrict__ dst = out  + (size_t)row * kSKV;

    if (l < s0 || l >= s1) {                  // outside the updated slice: pure copy
        #pragma unroll
        for (int i = 0; i < kRowPerT; ++i) {
            const int c = i * kThreads + t;
            reinterpret_cast<float4*>(dst)[c] =
                reinterpret_cast<const float4*>(src)[c];
        }
        return;
    }

    const int* __restrict__ irow =
        idx + ((size_t)b * (size_t)(s1 - s0) + (size_t)(l - s0)) * (size_t)kK;

#if HAVE_ASYNC_LDS
    // gfx1250 async DMA: stage the 8 KB index row into LDS (ASYNCcnt path),
    // overlapping with the mask-row loads below.
    #pragma unroll
    for (int i = 0; i < kIdxPerT; ++i) {
        const int c = i * kThreads + t;       // int4 chunk id
        __builtin_amdgcn_global_load_async_to_lds_b128(
            (AS1 v4i*)(irow + 4 * c),
            (AS3 v4i*)&lds_idx[4 * c], 0, 0);
    }
#endif

    // Pull the 32 KB mask row into VGPRs with b128 loads (overlaps async copy).
    float4 v[kRowPerT];
    #pragma unroll
    for (int i = 0; i < kRowPerT; ++i)
        v[i] = reinterpret_cast<const float4*>(src)[i * kThreads + t];

    // Zero the scatter bitmask and flags (disjoint from async target region).
    bits[t] = 0u;                             // kMaskWords == kThreads
    if (t == 0) flags = 0u;

#if HAVE_ASYNC_LDS
  #if __has_builtin(__builtin_amdgcn_s_wait_asynccnt)
    __builtin_amdgcn_s_wait_asynccnt(0);
  #else
    asm volatile("s_wait_asynccnt 0" ::: "memory");
  #endif
#endif
    __syncthreads();

    // Scatter: set one bit per hit column; duplicates are benign.
    unsigned sent = 0u, zero = 0u;
    auto proc = [&](int q) {
        if (q < 0) {
            sent = 1u;                        // sentinel: no store (fixup handles col 0)
        } else if (q < kSKV) {
            atomicOr(&bits[(unsigned)q >> 5], 1u << (q & 31));
            zero |= (q == 0) ? 1u : 0u;
        }
    };
    #pragma unroll
    for (int i = 0; i < kIdxPerT; ++i) {
        const int c = i * kThreads + t;
#if HAVE_ASYNC_LDS
        const int4 w = *reinterpret_cast<const int4*>(&lds_idx[4 * c]);
#else
        const int4 w = reinterpret_cast<const int4*>(irow)[c];
#endif
        proc(w.x); proc(w.y); proc(w.z); proc(w.w);
    }
    if (sent | zero) atomicOr(&flags, sent | (zero << 1));
    __syncthreads();

    const unsigned f     = flags;
    const bool     fixup = (f & 1u) && !(f & 2u);

    // Apply bitmask in registers and stream out with b128 stores.
    #pragma unroll
    for (int i = 0; i < kRowPerT; ++i) {
        const int      c    = i * kThreads + t;   // float4 chunk id; element base = 4c
        const unsigned word = bits[c >> 3];       // (4c)>>5 == c>>3
        const unsigned sh   = 4u * (c & 7);       // (4c)&31
        float4 o = v[i];
        if ((word >> (sh + 0)) & 1u) o.x = 0.0f;
        if ((word >> (sh + 1)) & 1u) o.y = 0.0f;
        if ((word >> (sh + 2)) & 1u) o.z = 0.0f;
        if ((word >> (sh + 3)) & 1u) o.w = 0.0f;
        if (c == 0 && fixup) o.x = -__builtin_inff();
        reinterpret_cast<float4*>(dst)[c] = o;
    }
}

extern "C" void kernel_launch(void* const* d_in, const int* in_sizes, int n_in,
                              void* d_out, int out_size, void* d_ws, size_t ws_size,
                              hipStream_t stream) {
    const float* mask = (const float*)d_in[0];
    const int*   idx  = (const int*)d_in[1];
    const int*   s0   = (const int*)d_in[2];
    const int*   s1   = (const int*)d_in[3];
    float*       out  = (float*)d_out;

    const dim3 grid(kB * kSQ);
    const dim3 block(kThreads);
    dsa_scatter_row_kernel<<<grid, block, 0, stream>>>(mask, idx, s0, s1, out);
}